// LabelClusterLoss_48163763257993
// MI455X (gfx1250) — compile-verified
//
#include <hip/hip_runtime.h>

// LabelClusterLoss for MI455X (gfx1250):
//   E is [C0*T, C1] = [65536, 1024] fp32 (256 MB). Pure bandwidth problem
//   (~11us floor at 23.3 TB/s). One streaming pass computes all sufficient
//   statistics; row-sum reduction is expressed as P(64x256) x ones(256) via
//   chained V_WMMA_F32_16X16X4_F32; data movement uses gfx1250 async
//   global->LDS double buffering (ASYNCcnt). No atomics -> deterministic.

#define TDIM 128
#define NC0  512
#define NC1  1024

typedef float v2f __attribute__((ext_vector_type(2)));
typedef float v8f __attribute__((ext_vector_type(8)));

// ---- workspace layout (floats) ----
#define OFF_CPART 0u                       // [8][128][1024] column-sum partials
#define N_CPART   (8u * 128u * 1024u)
#define OFF_S2    (N_CPART)                // [1024] per-block sum of squares
#define OFF_SR2   (OFF_S2 + 1024u)        // [1024] per-block sum of rowsum^2
#define OFF_SC2   (OFF_SR2 + 1024u)       // [128]  per-t sum of colsum^2
#define OFF_GT    (OFF_SC2 + 128u)        // [128]  per-t total sum G[t]

__device__ __forceinline__ float waveReduceSum(float v) {
#pragma unroll
  for (int off = 16; off > 0; off >>= 1) v += __shfl_xor(v, off, 32);
  return v;
}

// ---- gfx1250 async global->LDS (tracked by ASYNCcnt) ----
template <int N>
__device__ __forceinline__ void waitAsync() {
  asm volatile("s_wait_asynccnt %0" :: "n"(N) : "memory");
}

__device__ __forceinline__ void asyncLoad16(unsigned ldsByteOff,
                                            const float* sBase,
                                            unsigned vByteOff) {
  // GVS mode: lds-dest VGPR, 32-bit voffset VGPR, 64-bit SGPR base
  asm volatile("global_load_async_to_lds_b128 %0, %1, %2"
               :: "v"(ldsByteOff), "v"(vByteOff), "s"(sBase)
               : "memory");
}

#if __has_builtin(__builtin_amdgcn_wmma_f32_16x16x4_f32)
#define HAVE_WMMA_F32X4 1
#else
#define HAVE_WMMA_F32X4 0
#endif

// ================== kernel 1: streaming statistics ==================
// Block (t, ic): t = bid & 127, ic = bid >> 7. Owns 64 rows r = (ic*64+il)*128+t
// (all share t), all 1024 columns. Thread tid owns columns 4*tid..4*tid+3.
__global__ __launch_bounds__(256)
void lcl_k1_stats(const float* __restrict__ E, float* __restrict__ ws) {
  __shared__ float stage[2][4][1024];   // 32 KB double-buffered async stage
  __shared__ float P[64][258];          // row-sum partials, padded stride
  __shared__ float tileSR2[4];
  __shared__ float redbuf[8];

  const int tid  = threadIdx.x;
  const int wid  = tid >> 5;
  const int lane = tid & 31;
  const int bid  = blockIdx.x;
  const int t    = bid & (TDIM - 1);
  const int ic   = bid >> 7;

  const unsigned vByteOff = (unsigned)tid * 16u;

  auto rowPtr = [&](int il) -> const float* {
    const size_t r = (size_t)(ic * 64 + il) * TDIM + (size_t)t;
    return E + r * (size_t)NC1;
  };
  auto issueChunk = [&](int ch) {
    const int buf = ch & 1;
#pragma unroll
    for (int q = 0; q < 4; ++q) {
      const unsigned ldsOff =
          (unsigned)(size_t)(void*)(&stage[buf][q][0]) + vByteOff;
      asyncLoad16(ldsOff, rowPtr(ch * 4 + q), vByteOff);
    }
  };

  float sq = 0.f;
  float cs0 = 0.f, cs1 = 0.f, cs2 = 0.f, cs3 = 0.f;

  issueChunk(0);
  for (int ch = 0; ch < 16; ++ch) {
    if (ch + 1 < 16) { issueChunk(ch + 1); waitAsync<4>(); }
    else             { waitAsync<0>(); }
    const int buf = ch & 1;
#pragma unroll
    for (int q = 0; q < 4; ++q) {
      const float4 v = *(const float4*)&stage[buf][q][tid * 4];
      sq  += v.x * v.x + v.y * v.y + v.z * v.z + v.w * v.w;
      cs0 += v.x; cs1 += v.y; cs2 += v.z; cs3 += v.w;
      P[ch * 4 + q][tid] = (v.x + v.y) + (v.z + v.w);  // thread's row partial
    }
  }

  // column-sum partials -> ws (pure stores, no init needed)
  {
    float4 cs; cs.x = cs0; cs.y = cs1; cs.z = cs2; cs.w = cs3;
    const size_t idx =
        (size_t)OFF_CPART + ((size_t)ic * 128 + t) * 1024 + (size_t)tid * 4;
    *(float4*)&ws[idx] = cs;
  }

  // S2 partial: wave reduce, stash per-wave
  {
    const float wsum = waveReduceSum(sq);
    if (lane == 0) redbuf[wid] = wsum;
  }
  __syncthreads();  // P and redbuf visible

  // Row sums via WMMA: R(64) = P(64x256) x ones(256); waves 0..3, 16 rows each.
  if (wid < 4) {
    const int row = wid * 16 + (lane & 15);
    const int cb  = (lane >> 4) * 2;  // f32 A-layout: lanes 16-31 hold K=2,3
    float s = 0.f;
#if HAVE_WMMA_F32X4
    v8f acc = {0.f, 0.f, 0.f, 0.f, 0.f, 0.f, 0.f, 0.f};
    v2f bOnes; bOnes[0] = 1.f; bOnes[1] = 1.f;
#pragma unroll 8
    for (int k0 = 0; k0 < 256; k0 += 4) {
      const v2f a = *(const v2f*)&P[row][k0 + cb];
      acc = __builtin_amdgcn_wmma_f32_16x16x4_f32(
          false, a, false, bOnes, (short)0, acc, false, false);
    }
    // All 16 D columns equal R[row-tile] -> sum(D^2) over lanes/regs = 16*SR2
#pragma unroll
    for (int c = 0; c < 8; ++c) s += acc[c] * acc[c];
    s = waveReduceSum(s) * (1.0f / 16.0f);
#else
    // Fallback (no WMMA builtin): lane sums its half-K strip, pair lanes.
    float pr = 0.f;
    for (int k0 = 0; k0 < 256; k0 += 4)
      pr += P[row][k0 + cb] + P[row][k0 + cb + 1];
    pr += __shfl_xor(pr, 16, 32);     // lanes l and l+16 now hold full R
    s = waveReduceSum(pr * pr) * 0.5f;  // each row counted twice
#endif
    if (lane == 0) tileSR2[wid] = s;
  }
  __syncthreads();

  if (tid == 0) {
    float s2 = 0.f;
#pragma unroll
    for (int i = 0; i < 8; ++i) s2 += redbuf[i];
    ws[OFF_S2 + bid]  = s2;
    ws[OFF_SR2 + bid] = tileSR2[0] + tileSR2[1] + tileSR2[2] + tileSR2[3];
  }
}

// ================== kernel 2: fold column-sum partials ==================
__global__ __launch_bounds__(256)
void lcl_k2_colstats(float* __restrict__ ws) {
  const int t   = blockIdx.x;
  const int tid = threadIdx.x;
  __shared__ float bufA[8];
  __shared__ float bufB[8];

  float sc2 = 0.f, g = 0.f;
#pragma unroll
  for (int jj = 0; jj < 4; ++jj) {
    const int j = tid + jj * 256;
    float c = 0.f;
#pragma unroll
    for (int ic = 0; ic < 8; ++ic)
      c += ws[(size_t)OFF_CPART + ((size_t)ic * 128 + t) * 1024 + j];
    sc2 += c * c;
    g   += c;
  }
  sc2 = waveReduceSum(sc2);
  g   = waveReduceSum(g);
  const int wid = tid >> 5, lane = tid & 31;
  if (lane == 0) { bufA[wid] = sc2; bufB[wid] = g; }
  __syncthreads();
  if (tid == 0) {
    float a = 0.f, b = 0.f;
#pragma unroll
    for (int i = 0; i < 8; ++i) { a += bufA[i]; b += bufB[i]; }
    ws[OFF_SC2 + t] = a;
    ws[OFF_GT  + t] = b;
  }
}

// ================== kernel 3: finalize loss ==================
__global__ __launch_bounds__(256)
void lcl_k3_final(const float* __restrict__ ws, float* __restrict__ out) {
  const int tid = threadIdx.x;
  __shared__ float buf[8];

  float s2 = 0.f, sr2 = 0.f, sc2 = 0.f, sg2 = 0.f;
  for (int i = tid; i < 1024; i += 256) {
    s2  += ws[OFF_S2 + i];
    sr2 += ws[OFF_SR2 + i];
  }
  if (tid < 128) {
    sc2 = ws[OFF_SC2 + tid];
    const float gt = ws[OFF_GT + tid];
    sg2 = gt * gt;
  }

  auto bred = [&](float v) -> float {
    v = waveReduceSum(v);
    __syncthreads();
    if ((tid & 31) == 0) buf[tid >> 5] = v;
    __syncthreads();
    float r = 0.f;
    if (tid == 0) {
#pragma unroll
      for (int i = 0; i < 8; ++i) r += buf[i];
    }
    return r;  // valid on thread 0
  };
  s2 = bred(s2); sr2 = bred(sr2); sc2 = bred(sc2); sg2 = bred(sg2);

  if (tid == 0) {
    const float fc0 = (float)NC0, fc1 = (float)NC1;
    const float intra  = s2 - sr2 / fc1;
    const float inter  = (sr2 - sg2 / fc0) / fc1;
    const float dintra = s2 - sc2 / fc0;
    const float dinter = (sc2 - sg2 / fc1) / fc0;
    out[0] = 0.5f * (intra / inter + dintra / dinter) / (fc0 * fc1);
  }
}

extern "C" void kernel_launch(void* const* d_in, const int* in_sizes, int n_in,
                              void* d_out, int out_size, void* d_ws, size_t ws_size,
                              hipStream_t stream) {
  const float* E = (const float*)d_in[0];  // [65536, 1024] fp32
  float* ws  = (float*)d_ws;               // needs ~4.01 MB
  float* out = (float*)d_out;              // 1 fp32

  lcl_k1_stats   <<<1024, 256, 0, stream>>>(E, ws);
  lcl_k2_colstats<<<128,  256, 0, stream>>>(ws);
  lcl_k3_final   <<<1,    256, 0, stream>>>(ws, out);

  (void)in_sizes; (void)n_in; (void)out_size; (void)ws_size;
}